// DQNModel_11665131176635
// MI455X (gfx1250) — compile-verified
//
#include <hip/hip_runtime.h>
#include <hip/hip_bf16.h>

typedef __attribute__((ext_vector_type(2))) float v2f;
typedef __attribute__((ext_vector_type(8))) float v8f;

#define NN 50000
#define NE 400000
#define NG 64
#define DD 128
#define NA 40
#define LDS_STRIDE 132   // 128 + 4 pad -> conflict-free fragment reads

// ---------------------------------------------------------------------------
// Segment-sum pooling over sorted graph ids: OUT[g][d] += sum of F[n][d].
// Each block handles 128 nodes; thread = (column, node-substream). Running
// accumulator flushed via atomicAdd only on segment change (ids are sorted).
// ---------------------------------------------------------------------------
__global__ __launch_bounds__(256)
void pool_kernel(const float* __restrict__ F, const int* __restrict__ G,
                 float* __restrict__ OUT, int nNodes)
{
    const int col  = threadIdx.x & 127;
    const int sub  = threadIdx.x >> 7;           // 0 or 1
    const int start = blockIdx.x * 128;
    int end = start + 128; if (end > nNodes) end = nNodes;

    float acc = 0.0f;
    int cur = -1;
    for (int n = start + sub; n < end; n += 2) {
        int g = G[n];
        float v = F[(size_t)n * DD + col];
        if (g != cur) {
            if (cur >= 0) atomicAdd(&OUT[cur * DD + col], acc);
            cur = g; acc = v;
        } else {
            acc += v;
        }
    }
    if (cur >= 0) atomicAdd(&OUT[cur * DD + col], acc);
}

// ---------------------------------------------------------------------------
// Edge scatter: AGG[dst] += H[src]. 32 threads/edge, float4 per thread.
// 400000*32 = 12.8M threads exactly; node features are L2-resident (25.6MB).
// ---------------------------------------------------------------------------
__global__ __launch_bounds__(256)
void edge_agg_kernel(const float* __restrict__ H, const int* __restrict__ SRC,
                     const int* __restrict__ DST, float* __restrict__ AGG)
{
    int tid = blockIdx.x * 256 + threadIdx.x;
    int e = tid >> 5;
    if (e >= NE) return;
    int q = (tid & 31) * 4;
    int s = SRC[e];
    int d = DST[e];
    const float4 v = *(const float4*)(H + (size_t)s * DD + q);
    float* base = AGG + (size_t)d * DD + q;
    atomicAdd(base + 0, v.x);
    atomicAdd(base + 1, v.y);
    atomicAdd(base + 2, v.z);
    atomicAdd(base + 3, v.w);
}

// ---------------------------------------------------------------------------
// CTXL = CTX + HG  (per-layer broadcast context), 64*128 elements.
// ---------------------------------------------------------------------------
__global__ __launch_bounds__(256)
void ctx_add_kernel(const float* __restrict__ CTX, const float* __restrict__ HG,
                    float* __restrict__ CTXL)
{
    int i = blockIdx.x * 256 + threadIdx.x;
    if (i < NG * DD) CTXL[i] = CTX[i] + HG[i];
}

// ---------------------------------------------------------------------------
// Fused GGIN GEMM: OUT = relu((H + AGG + CTXL[gid]) @ W + bias)
// fp32 WMMA 16x16x4. One wave per 16-column tile (8 waves = 128 cols).
// B fragments (all 32 K-steps) held in registers for the whole kernel.
// A tile (16x128) staged in padded LDS with the context add fused in.
// A sched_barrier(0) between the fragment loads and the MAC chain forces
// all 16 ds_load_2addr_b64 to issue first (single DScnt drain), then the
// 32 WMMAs run back-to-back on the matrix pipe with no memory waits.
// ---------------------------------------------------------------------------
__global__ __launch_bounds__(256)
void ggin_gemm_wmma(const float* __restrict__ H, const float* __restrict__ AGG,
                    const float* __restrict__ CTXL, const int* __restrict__ GID,
                    const float* __restrict__ W, const float* __restrict__ Bv,
                    float* __restrict__ OUT, int nTiles)
{
    __shared__ float tile[16 * LDS_STRIDE];

    const int t     = threadIdx.x;
    const int wave  = t >> 5;          // 0..7 -> column tile
    const int lane  = t & 31;
    const int lhalf = lane & 15;
    const int hi    = lane >> 4;       // 0: K/M low pair, 1: high pair

    // Preload this wave's B fragments: B[4x16] per K-step s.
    // Layout: v0 = B[K=hi*2][N=lhalf], v1 = B[K=hi*2+1][N=lhalf].
    const int n0 = wave * 16 + lhalf;  // global output column of this lane
    v2f bfrag[32];
#pragma unroll
    for (int s = 0; s < 32; ++s) {
        int krow = s * 4 + hi * 2;
        bfrag[s].x = W[(size_t)krow * DD + n0];
        bfrag[s].y = W[(size_t)(krow + 1) * DD + n0];
    }
    const float bias = Bv[n0];

    const float* aBase = &tile[lhalf * LDS_STRIDE + hi * 2];

    for (int tileIdx = blockIdx.x; tileIdx < nTiles; tileIdx += gridDim.x) {
        const int row0 = tileIdx * 16;

        // Stage X = H + AGG + CTXL[gid] into LDS (512 float4s, 2 per thread)
#pragma unroll
        for (int it = 0; it < 2; ++it) {
            int idx = t + it * 256;            // 0..511
            int r = idx >> 5;                  // 0..15
            int c = (idx & 31) * 4;            // 0..124
            int g = GID[row0 + r];
            const float4 hv = *(const float4*)(H    + (size_t)(row0 + r) * DD + c);
            const float4 av = *(const float4*)(AGG  + (size_t)(row0 + r) * DD + c);
            const float4 cv = *(const float4*)(CTXL + (size_t)g * DD + c);
            float* dst = &tile[r * LDS_STRIDE + c];
            dst[0] = hv.x + av.x + cv.x;
            dst[1] = hv.y + av.y + cv.y;
            dst[2] = hv.z + av.z + cv.z;
            dst[3] = hv.w + av.w + cv.w;
        }
        __syncthreads();

        // Pull all 32 A-fragments into registers; the sched_barrier pins the
        // loads before the MAC chain so the WMMAs issue densely.
        v2f afrag[32];
#pragma unroll
        for (int s = 0; s < 32; ++s) {
            afrag[s] = *(const v2f*)(aBase + s * 4);
        }

        __builtin_amdgcn_sched_barrier(0);

        v8f acc = {};
#pragma unroll
        for (int s = 0; s < 32; ++s) {
            acc = __builtin_amdgcn_wmma_f32_16x16x4_f32(
                false, afrag[s], false, bfrag[s], (short)0, acc, false, false);
        }

        // Store with bias + relu. C/D layout: VGPR r -> M = r (+8 for hi lanes)
#pragma unroll
        for (int r = 0; r < 8; ++r) {
            int row = row0 + hi * 8 + r;
            float v = acc[r] + bias;
            v = v > 0.0f ? v : 0.0f;
            OUT[(size_t)row * DD + n0] = v;
        }
        __syncthreads();
    }
}

// ---------------------------------------------------------------------------
// fc1: HFC[64x128] = relu(HG @ W + b). Tiny (2 MFLOP) -> plain dot kernel.
// ---------------------------------------------------------------------------
__global__ __launch_bounds__(128)
void fc1_kernel(const float* __restrict__ HG, const float* __restrict__ W,
                const float* __restrict__ Bv, float* __restrict__ HFC)
{
    int r = blockIdx.x;       // 0..63
    int c = threadIdx.x;      // 0..127
    float acc = Bv[c];
    const float* hrow = HG + (size_t)r * DD;
#pragma unroll 8
    for (int k = 0; k < DD; ++k) acc += hrow[k] * W[(size_t)k * DD + c];
    HFC[(size_t)r * DD + c] = acc > 0.0f ? acc : 0.0f;
}

// ---------------------------------------------------------------------------
// fc2: OUT[64x40] = HFC @ W + b
// ---------------------------------------------------------------------------
__global__ __launch_bounds__(64)
void fc2_kernel(const float* __restrict__ HFC, const float* __restrict__ W,
                const float* __restrict__ Bv, float* __restrict__ OUT)
{
    int r = blockIdx.x;       // 0..63
    int c = threadIdx.x;      // 0..63 (only <40 used)
    if (c >= NA) return;
    float acc = Bv[c];
    const float* hrow = HFC + (size_t)r * DD;
#pragma unroll 8
    for (int k = 0; k < DD; ++k) acc += hrow[k] * W[(size_t)k * NA + c];
    OUT[(size_t)r * NA + c] = acc;
}

extern "C" void kernel_launch(void* const* d_in, const int* in_sizes, int n_in,
                              void* d_out, int out_size, void* d_ws, size_t ws_size,
                              hipStream_t stream) {
    const float* inputs   = (const float*)d_in[0];
    const int*   src      = (const int*)  d_in[1];
    const int*   dst      = (const int*)  d_in[2];
    const int*   gids     = (const int*)  d_in[3];
    const float* initF    = (const float*)d_in[4];
    const int*   initG    = (const int*)  d_in[5];
    const float* leadF    = (const float*)d_in[6];
    const int*   leadG    = (const int*)  d_in[7];
    const float* W1 = (const float*)d_in[8];
    const float* b1 = (const float*)d_in[9];
    const float* W2 = (const float*)d_in[10];
    const float* b2 = (const float*)d_in[11];
    const float* W3 = (const float*)d_in[12];
    const float* b3 = (const float*)d_in[13];
    const float* fc1_W = (const float*)d_in[14];
    const float* fc1_b = (const float*)d_in[15];
    const float* fc2_W = (const float*)d_in[16];
    const float* fc2_b = (const float*)d_in[17];
    float* out = (float*)d_out;

    // Workspace layout (floats)
    const size_t NND = (size_t)NN * DD;     // 6.4M floats
    float* hA   = (float*)d_ws;
    float* hB   = hA + NND;
    float* agg  = hB + NND;
    float* ctx  = agg + NND;                // [64*128]
    float* hg   = ctx + NG * DD;
    float* ctxl = hg + NG * DD;
    float* hfc  = ctxl + NG * DD;

    const int poolGrid = (NN + 127) / 128;          // 391
    const int edgeGrid = (NE * 32) / 256;           // 50000
    const int nTiles   = NN / 16;                   // 3125
    const int gemmGrid = 1024;

    // ctx = pool(init) + pool(lead)   (both accumulate into same buffer)
    hipMemsetAsync(ctx, 0, (size_t)NG * DD * sizeof(float), stream);
    pool_kernel<<<poolGrid, 256, 0, stream>>>(initF, initG, ctx, NN);
    pool_kernel<<<poolGrid, 256, 0, stream>>>(leadF, leadG, ctx, NN);

    const float* layerIn[3]  = {inputs, hA, hB};
    float*       layerOut[3] = {hA, hB, hA};
    const float* Ws[3] = {W1, W2, W3};
    const float* bs[3] = {b1, b2, b3};

    for (int l = 0; l < 3; ++l) {
        const float* hin = layerIn[l];
        // per-layer context: ctxl = ctx + pool(hin)
        hipMemsetAsync(hg, 0, (size_t)NG * DD * sizeof(float), stream);
        pool_kernel<<<poolGrid, 256, 0, stream>>>(hin, gids, hg, NN);
        ctx_add_kernel<<<(NG * DD + 255) / 256, 256, 0, stream>>>(ctx, hg, ctxl);
        // neighbor aggregation
        hipMemsetAsync(agg, 0, NND * sizeof(float), stream);
        edge_agg_kernel<<<edgeGrid, 256, 0, stream>>>(hin, src, dst, agg);
        // fused (h + agg + ctx) @ W + b, relu -> WMMA fp32
        ggin_gemm_wmma<<<gemmGrid, 256, 0, stream>>>(hin, agg, ctxl, gids,
                                                     Ws[l], bs[l], layerOut[l], nTiles);
    }

    // final readout
    hipMemsetAsync(hg, 0, (size_t)NG * DD * sizeof(float), stream);
    pool_kernel<<<poolGrid, 256, 0, stream>>>(layerOut[2], gids, hg, NN);
    fc1_kernel<<<NG, 128, 0, stream>>>(hg, fc1_W, fc1_b, hfc);
    fc2_kernel<<<NG, 64, 0, stream>>>(hfc, fc2_W, fc2_b, out);
}